// AdaptiveTripletLoss_27118423507256
// MI455X (gfx1250) — compile-verified
//
#include <hip/hip_runtime.h>
#include <math.h>

typedef __attribute__((ext_vector_type(2))) float v2f;
typedef __attribute__((ext_vector_type(8))) float v8f;

#define N_ROWS 4096
#define DIM 2048
#define CHUNK 512
#define NUM_CHUNKS (N_ROWS / CHUNK)
#define MARGIN 0.3f
#define EPS 1e-6f
#define LDS_PITCH 36                 // 32 K-floats + 4 pad: conflict-free frag reads, 16B-aligned
#define TILE_FLOATS (64 * LDS_PITCH)

// ---------------------------------------------------------------------------
// threefry2x32 (exact JAX PRNG replication)
// ---------------------------------------------------------------------------
__device__ __forceinline__ unsigned int rotl32(unsigned int x, int d) {
    return (x << d) | (x >> (32 - d));
}

__device__ __forceinline__ void threefry2x32(unsigned int k0, unsigned int k1,
                                             unsigned int c0, unsigned int c1,
                                             unsigned int* o0, unsigned int* o1) {
    unsigned int ks2 = k0 ^ k1 ^ 0x1BD11BDAu;
    unsigned int x0 = c0 + k0;
    unsigned int x1 = c1 + k1;
    const int ra[4] = {13, 15, 26, 6};
    const int rb[4] = {17, 29, 16, 24};
#define TF_R4(R) { for (int r = 0; r < 4; r++) { x0 += x1; x1 = rotl32(x1, R[r]); x1 ^= x0; } }
    TF_R4(ra); x0 += k1;  x1 += ks2 + 1u;
    TF_R4(rb); x0 += ks2; x1 += k0 + 2u;
    TF_R4(ra); x0 += k0;  x1 += k1 + 3u;
    TF_R4(rb); x0 += k1;  x1 += ks2 + 4u;
    TF_R4(ra); x0 += ks2; x1 += k0 + 5u;
#undef TF_R4
    *o0 = x0; *o1 = x1;
}

// u = jax.random.uniform(keys[q], (4096,))[row] where keys = split(key(42), 6)
__device__ __forceinline__ float jax_uniform_q_row(int q, int row) {
    unsigned int flat[12];
    for (int j = 0; j < 6; j++) {
        unsigned int a, b;
        threefry2x32(0u, 42u, (unsigned int)j, (unsigned int)(j + 6), &a, &b);
        flat[j] = a; flat[6 + j] = b;
    }
    unsigned int kk0 = flat[2 * q], kk1 = flat[2 * q + 1];
    unsigned int b0, b1, bits;
    if (row < 2048) {
        threefry2x32(kk0, kk1, (unsigned int)row, (unsigned int)(row + 2048), &b0, &b1);
        bits = b0;
    } else {
        threefry2x32(kk0, kk1, (unsigned int)(row - 2048), (unsigned int)row, &b0, &b1);
        bits = b1;
    }
    return __uint_as_float((bits >> 9) | 0x3F800000u) - 1.0f;
}

// ---------------------------------------------------------------------------
// Kernel 1: squared row norms (float4 streaming)
// ---------------------------------------------------------------------------
__global__ __launch_bounds__(256) void sqnorm_kernel(const float* __restrict__ emb,
                                                     float* __restrict__ sq) {
    __shared__ float red[256];
    int row = blockIdx.x, tid = threadIdx.x;
    const float4* e4 = (const float4*)(emb + (size_t)row * DIM);
    float s = 0.0f;
    for (int d = tid; d < DIM / 4; d += 256) {
        float4 v = e4[d];
        s += v.x * v.x + v.y * v.y + v.z * v.z + v.w * v.w;
    }
    red[tid] = s;
    __syncthreads();
    for (int off = 128; off > 0; off >>= 1) {
        if (tid < off) red[tid] += red[tid + off];
        __syncthreads();
    }
    if (tid == 0) sq[row] = red[0];
}

// ---------------------------------------------------------------------------
// Kernel 2: WMMA fp32 GEMM (E x E^T) fused with distance epilogue.
// Block = 256 threads (8 wave32). 64x64 output tile; each wave owns a 16x32
// strip (two 16x16 WMMA accumulators, A-fragment reused across both).
// Double-buffered GLOBAL_LOAD_ASYNC_TO_LDS_B128 staging: next tile's DMA is
// issued before compute, s_wait_asynccnt 0x4 retires only the older group
// (per-wave in-order completion), overlapping DMA with the WMMA burst.
// ---------------------------------------------------------------------------
__global__ __launch_bounds__(256) void gemm_dist_kernel(const float* __restrict__ emb,
                                                        const float* __restrict__ sq,
                                                        float* __restrict__ dists,
                                                        int rowBase) {
    __shared__ float As[2][TILE_FLOATS];   // [buf][row 0..63][k 0..31]
    __shared__ float Bs[2][TILE_FLOATS];   // [buf][col 0..63][k 0..31]
    int tid   = threadIdx.x;
    int tileN = blockIdx.x * 64;
    int tileM = blockIdx.y * 64;
    int wave  = tid >> 5;
    int lane  = tid & 31;
    int h     = lane >> 4;     // lane half selects K pair per ISA fragment layout
    int n15   = lane & 15;     // M (A frag) / N (B frag)
    int wm = wave >> 1;        // 0..3 -> 16-row strip
    int wn = wave & 1;         // 0..1 -> 32-col strip

    // Per-thread staging assignment: 512 float4 chunks per matrix, 2 each.
    int r0 = tid >> 3;                 // rows 0..31 (and +32)
    int c4 = (tid & 7) << 2;           // k offset 0..28
    const unsigned int HALF = (unsigned int)(32 * LDS_PITCH * 4);   // +32 rows, bytes
    const unsigned int BUFB = (unsigned int)(TILE_FLOATS * 4);      // buffer stride, bytes
    unsigned int lA = (unsigned int)(size_t)(void*)(&As[0][r0 * LDS_PITCH + c4]);
    unsigned int lB = (unsigned int)(size_t)(void*)(&Bs[0][r0 * LDS_PITCH + c4]);
    const float* gArow0 = emb + (size_t)(rowBase + tileM + r0) * DIM + c4;
    const float* gArow1 = gArow0 + (size_t)32 * DIM;
    const float* gBrow0 = emb + (size_t)(tileN + r0) * DIM + c4;
    const float* gBrow1 = gBrow0 + (size_t)32 * DIM;

#define STAGE(k0, buf)                                                                           \
    {                                                                                            \
        unsigned int bo = (buf) ? BUFB : 0u;                                                     \
        unsigned long long a0 = (unsigned long long)(size_t)(gArow0 + (k0));                     \
        unsigned long long a1 = (unsigned long long)(size_t)(gArow1 + (k0));                     \
        unsigned long long b0 = (unsigned long long)(size_t)(gBrow0 + (k0));                     \
        unsigned long long b1 = (unsigned long long)(size_t)(gBrow1 + (k0));                     \
        asm volatile("global_load_async_to_lds_b128 %0, %1, off" :: "v"(lA + bo), "v"(a0) : "memory"); \
        asm volatile("global_load_async_to_lds_b128 %0, %1, off" :: "v"(lA + bo + HALF), "v"(a1) : "memory"); \
        asm volatile("global_load_async_to_lds_b128 %0, %1, off" :: "v"(lB + bo), "v"(b0) : "memory"); \
        asm volatile("global_load_async_to_lds_b128 %0, %1, off" :: "v"(lB + bo + HALF), "v"(b1) : "memory"); \
    }

    v8f acc0 = {};
    v8f acc1 = {};
    int cur = 0;
    STAGE(0, 0);
    for (int k0 = 0; k0 < DIM; k0 += 32) {
        if (k0 + 32 < DIM) {
            STAGE(k0 + 32, cur ^ 1);
            asm volatile("s_wait_asynccnt 0x4" ::: "memory");   // retire current buffer's loads only
        } else {
            asm volatile("s_wait_asynccnt 0x0" ::: "memory");
        }
        __syncthreads();   // all waves' current-buffer DMA complete

        const float* ap  = &As[cur][(wm * 16 + n15) * LDS_PITCH];
        const float* bp0 = &Bs[cur][(wn * 32 + n15) * LDS_PITCH];
        const float* bp1 = bp0 + 16 * LDS_PITCH;
        #pragma unroll
        for (int kb = 0; kb < 32; kb += 4) {
            v2f a, b0, b1;
            a.x  = ap[kb + 2 * h];
            a.y  = ap[kb + 2 * h + 1];
            b0.x = bp0[kb + 2 * h];
            b0.y = bp0[kb + 2 * h + 1];
            b1.x = bp1[kb + 2 * h];
            b1.y = bp1[kb + 2 * h + 1];
            acc0 = __builtin_amdgcn_wmma_f32_16x16x4_f32(false, a, false, b0,
                                                         (short)0, acc0, false, false);
            acc1 = __builtin_amdgcn_wmma_f32_16x16x4_f32(false, a, false, b1,
                                                         (short)0, acc1, false, false);
        }
        __syncthreads();   // everyone done reading before next iteration overwrites this buffer
        cur ^= 1;
    }
#undef STAGE

    // Epilogue: C VGPR r holds (M=r, N=lane) lanes 0-15, (M=r+8, N=lane-16) lanes 16-31
    int col0 = tileN + wn * 32 + n15;
    int col1 = col0 + 16;
    float sqJ0 = sq[col0];
    float sqJ1 = sq[col1];
    #pragma unroll
    for (int r = 0; r < 8; r++) {
        int lr = tileM + wm * 16 + r + 8 * h;
        float sqI = sq[rowBase + lr];
        float d20 = sqI + sqJ0 - 2.0f * acc0[r];
        float d21 = sqI + sqJ1 - 2.0f * acc1[r];
        dists[(size_t)lr * N_ROWS + col0] = sqrtf(fmaxf(d20, 0.0f));
        dists[(size_t)lr * N_ROWS + col1] = sqrtf(fmaxf(d21, 0.0f));
    }
}

// ---------------------------------------------------------------------------
// Kernel 3: per-row masked order-statistic selection via 8-bit radix-select
// over the row cached in LDS; parallel block scans (no serial thread-0 loops).
// Stable tie-break matches jnp.argsort order.
// ---------------------------------------------------------------------------
__global__ __launch_bounds__(256) void select_kernel(const float* __restrict__ dists,
                                                     const int* __restrict__ labels,
                                                     int rowBase,
                                                     int* __restrict__ posIdx,
                                                     int* __restrict__ negIdx,
                                                     int* __restrict__ validF) {
    __shared__ unsigned int keysP[N_ROWS];
    __shared__ unsigned int keysN[N_ROWS];
    __shared__ unsigned int hist[256];
    __shared__ unsigned int red[256];
    __shared__ unsigned int bc[4];   // [0]=rank, [1]=prefix/key, [3]=result idx
    __shared__ int counts[2];

    int tid  = threadIdx.x;
    int lrow = blockIdx.x;
    int grow = rowBase + lrow;
    int myLabel = labels[grow];
    const float* drow = dists + (size_t)lrow * N_ROWS;
    int base = tid * 16;

    unsigned int cp = 0, cn = 0;
    for (int e = 0; e < 16; e++) {
        int j = base + e;
        float d = drow[j];
        int lj = labels[j];
        bool pos = (lj == myLabel) && (j != grow);
        bool neg = (lj != myLabel);
        keysP[j] = pos ? __float_as_uint(d) : 0xFFFFFFFFu;  // dists >= 0: bits order-preserving
        keysN[j] = neg ? __float_as_uint(d) : 0xFFFFFFFFu;
        cp += pos ? 1u : 0u;
        cn += neg ? 1u : 0u;
    }
    red[tid] = cp | (cn << 16);      // packed: each field total <= 4096, no carry
    __syncthreads();
    for (int off = 128; off > 0; off >>= 1) {
        if (tid < off) red[tid] += red[tid + off];
        __syncthreads();
    }
    if (tid == 0) {
        unsigned int packed = red[0];
        counts[0] = (int)(packed & 0xFFFFu);
        counts[1] = (int)(packed >> 16);
        validF[grow] = ((packed & 0xFFFFu) > 0u && (packed >> 16) > 0u) ? 1 : 0;
    }
    __syncthreads();

    const float ratios[3] = {0.3f, 0.4f, 0.3f};
    for (int q = 0; q < 6; q++) {
        const unsigned int* keys = (q < 3) ? keysP : keysN;
        int count = (q < 3) ? counts[0] : counts[1];
        int cnt = (int)floorf((float)count * ratios[q % 3]);
        if (cnt < 1) cnt = 1;
        float u = jax_uniform_q_row(q, grow);
        int j = (int)(u * (float)cnt);
        if (j > cnt - 1) j = cnt - 1;
        if (j < 0) j = 0;

        if (tid == 0) { bc[0] = (unsigned int)j; bc[1] = 0u; }
        __syncthreads();
        for (int shift = 24; shift >= 0; shift -= 8) {
            unsigned int prefix = bc[1];
            unsigned int rank   = bc[0];
            hist[tid] = 0u;
            __syncthreads();
            unsigned int hiMask = (shift == 24) ? 0u : (0xFFFFFFFFu << (shift + 8));
            for (int e = 0; e < 16; e++) {
                unsigned int k = keys[base + e];
                if ((k & hiMask) == prefix) atomicAdd(&hist[(k >> shift) & 255u], 1u);
            }
            __syncthreads();
            unsigned int v = hist[tid];
            red[tid] = v;
            __syncthreads();
            // Hillis-Steele inclusive scan over 256 bins
            for (int off = 1; off < 256; off <<= 1) {
                unsigned int t = (tid >= off) ? red[tid - off] : 0u;
                __syncthreads();
                red[tid] += t;
                __syncthreads();
            }
            unsigned int incl = red[tid];
            unsigned int excl = incl - v;
            if (excl <= rank && rank < incl) {           // exactly one digit bin matches
                bc[0] = rank - excl;
                bc[1] = prefix | ((unsigned int)tid << shift);
            }
            __syncthreads();
        }
        unsigned int K = bc[1], rloc = bc[0];
        // stable tie-break: rloc-th (by original index) element with key==K
        unsigned int eq = 0;
        for (int e = 0; e < 16; e++) if (keys[base + e] == K) eq++;
        red[tid] = eq;
        __syncthreads();
        for (int off = 1; off < 256; off <<= 1) {
            unsigned int t = (tid >= off) ? red[tid - off] : 0u;
            __syncthreads();
            red[tid] += t;
            __syncthreads();
        }
        unsigned int incl = red[tid];
        unsigned int excl = incl - eq;
        if (excl <= rloc && rloc < incl) {
            unsigned int want = rloc - excl, c = 0;
            for (int e = 0; e < 16; e++) {
                if (keys[base + e] == K) {
                    if (c == want) { bc[3] = (unsigned int)(base + e); break; }
                    c++;
                }
            }
        }
        __syncthreads();
        if (tid == 0) {
            int sel = (int)bc[3];
            if (q < 3) posIdx[grow * 3 + q] = sel;
            else       negIdx[grow * 3 + (q - 3)] = sel;
        }
        __syncthreads();
    }
}

// ---------------------------------------------------------------------------
// Kernel 4: triplet loss per row (norms recomputed with +EPS per component)
// ---------------------------------------------------------------------------
__global__ __launch_bounds__(256) void triplet_kernel(const float* __restrict__ emb,
                                                      const int* __restrict__ posIdx,
                                                      const int* __restrict__ negIdx,
                                                      const int* __restrict__ validF,
                                                      float* __restrict__ rowTri) {
    __shared__ float redA[256];
    __shared__ float redB[256];
    int row = blockIdx.x, tid = threadIdx.x;
    if (!validF[row]) {
        if (tid == 0) rowTri[row] = 0.0f;
        return;
    }
    const float4* a4 = (const float4*)(emb + (size_t)row * DIM);
    float total = 0.0f;
    for (int t = 0; t < 3; t++) {
        int p  = posIdx[row * 3 + t];
        int ng = negIdx[row * 3 + t];
        const float4* p4 = (const float4*)(emb + (size_t)p * DIM);
        const float4* n4 = (const float4*)(emb + (size_t)ng * DIM);
        float sap = 0.0f, san = 0.0f;
        for (int d = tid; d < DIM / 4; d += 256) {
            float4 av = a4[d], pv = p4[d], nv = n4[d];
            float dx, dy, dz, dw;
            dx = av.x - pv.x + EPS; dy = av.y - pv.y + EPS;
            dz = av.z - pv.z + EPS; dw = av.w - pv.w + EPS;
            sap += dx * dx + dy * dy + dz * dz + dw * dw;
            dx = av.x - nv.x + EPS; dy = av.y - nv.y + EPS;
            dz = av.z - nv.z + EPS; dw = av.w - nv.w + EPS;
            san += dx * dx + dy * dy + dz * dz + dw * dw;
        }
        redA[tid] = sap; redB[tid] = san;
        __syncthreads();
        for (int off = 128; off > 0; off >>= 1) {
            if (tid < off) { redA[tid] += redA[tid + off]; redB[tid] += redB[tid + off]; }
            __syncthreads();
        }
        if (tid == 0) {
            float dap = sqrtf(redA[0]);
            float dan = sqrtf(redB[0]);
            total += fmaxf(dap - dan + MARGIN, 0.0f);
        }
        __syncthreads();
    }
    if (tid == 0) rowTri[row] = total;
}

// ---------------------------------------------------------------------------
// Kernel 5: deterministic final reduction
// ---------------------------------------------------------------------------
__global__ __launch_bounds__(256) void final_kernel(const float* __restrict__ rowTri,
                                                    const int* __restrict__ validF,
                                                    float* __restrict__ out) {
    __shared__ float redS[256];
    __shared__ float redV[256];
    int tid = threadIdx.x;
    float s = 0.0f, v = 0.0f;
    for (int i = tid; i < N_ROWS; i += 256) {
        s += rowTri[i];
        v += (float)validF[i];
    }
    redS[tid] = s; redV[tid] = v;
    __syncthreads();
    for (int off = 128; off > 0; off >>= 1) {
        if (tid < off) { redS[tid] += redS[tid + off]; redV[tid] += redV[tid + off]; }
        __syncthreads();
    }
    if (tid == 0) out[0] = redS[0] / fmaxf(3.0f * redV[0], 1.0f);
}

// ---------------------------------------------------------------------------
extern "C" void kernel_launch(void* const* d_in, const int* in_sizes, int n_in,
                              void* d_out, int out_size, void* d_ws, size_t ws_size,
                              hipStream_t stream) {
    const float* emb   = (const float*)d_in[0];
    const int* labels  = (const int*)d_in[1];
    float* out = (float*)d_out;

    // workspace carve-up (~8.5 MB)
    float* dists  = (float*)d_ws;                         // CHUNK * 4096 floats
    float* sq     = dists + (size_t)CHUNK * N_ROWS;       // 4096 floats
    int*   posIdx = (int*)(sq + N_ROWS);                  // 4096*3 ints
    int*   negIdx = posIdx + N_ROWS * 3;                  // 4096*3 ints
    int*   validF = negIdx + N_ROWS * 3;                  // 4096 ints
    float* rowTri = (float*)(validF + N_ROWS);            // 4096 floats

    sqnorm_kernel<<<N_ROWS, 256, 0, stream>>>(emb, sq);

    for (int c = 0; c < NUM_CHUNKS; c++) {
        int rowBase = c * CHUNK;
        dim3 grid(N_ROWS / 64, CHUNK / 64);
        gemm_dist_kernel<<<grid, 256, 0, stream>>>(emb, sq, dists, rowBase);
        select_kernel<<<CHUNK, 256, 0, stream>>>(dists, labels, rowBase, posIdx, negIdx, validF);
    }

    triplet_kernel<<<N_ROWS, 256, 0, stream>>>(emb, posIdx, negIdx, validF, rowTri);
    final_kernel<<<1, 256, 0, stream>>>(rowTri, validF, out);
}